// EmbeddingToProbability_45689862094885
// MI455X (gfx1250) — compile-verified
//
#include <hip/hip_runtime.h>
#include <math.h>

// CDNA5 wave32 WMMA vector types
typedef float v2f __attribute__((ext_vector_type(2)));
typedef float v8f __attribute__((ext_vector_type(8)));

#define EPS_F 1e-16f
#define NCENT 64

// out[n,p] = exp( q_p + b_n + sum_c w[n,c]*e[c,p] )
//   q_p   = sum_c inv_s_c * e[c,p]^2
//   w[n,c]= -2 * inv_s_c * cent[n,c]
//   b_n   = sum_c inv_s_c * cent[n,c]^2
// Cross-term done as 64x4 @ 4x16 GEMM tiles via V_WMMA_F32_16X16X4_F32 (K=3 padded).
__global__ __launch_bounds__(256) void embed2prob_wmma_kernel(
    const float* __restrict__ emb,    // [3, P]
    const float* __restrict__ cent,   // [64, 3]
    const float* __restrict__ sigma,  // [3]
    float* __restrict__ out,          // [64, P]
    int P, int numTiles)
{
    __shared__ float sW0[NCENT], sW1[NCENT], sW2[NCENT], sBias[NCENT];
    __shared__ float sInv[3];

    const int tid = threadIdx.x;

    // --- per-block preamble: inv_s, w rows, bias into LDS ---
    if (tid < 3) {
        float sg = sigma[tid];
        sInv[tid] = 1.0f / (sg * sg * (-2.0f) + EPS_F);
    }
    __syncthreads();
    if (tid < NCENT) {
        float i0 = sInv[0], i1 = sInv[1], i2 = sInv[2];
        float c0 = cent[tid * 3 + 0];
        float c1 = cent[tid * 3 + 1];
        float c2 = cent[tid * 3 + 2];
        sW0[tid]  = -2.0f * i0 * c0;
        sW1[tid]  = -2.0f * i1 * c1;
        sW2[tid]  = -2.0f * i2 * c2;
        sBias[tid] = i0 * c0 * c0 + i1 * c1 * c1 + i2 * c2 * c2;
    }
    __syncthreads();

    const int lane = tid & 31;
    const int half = lane >> 4;   // 0: lanes 0-15 (K=0,1) ; 1: lanes 16-31 (K=2,3)
    const int lm   = lane & 15;   // N (pixel) / M (row) index within tile

    const float i0 = sInv[0], i1 = sInv[1], i2 = sInv[2];

    // --- loop-invariant A tiles: 16x4 f32, one per 16-centroid group ---
    // A layout: lanes 0-15 hold M=0..15 {K0,K1}; lanes 16-31 hold M=0..15 {K2,K3}
    v2f A[4];
#pragma unroll
    for (int g = 0; g < 4; ++g) {
        int r = g * 16 + lm;
        if (half == 0) { A[g].x = sW0[r]; A[g].y = sW1[r]; }
        else           { A[g].x = sW2[r]; A[g].y = 0.0f;  }
    }

    const int wavesPerBlock = blockDim.x >> 5;
    const int waveId     = blockIdx.x * wavesPerBlock + (tid >> 5);
    const int waveStride = gridDim.x * wavesPerBlock;

    const float* __restrict__ e0 = emb;
    const float* __restrict__ e1 = emb + (size_t)P;
    const float* __restrict__ e2 = emb + 2 * (size_t)P;

    for (int t = waveId; t < numTiles; t += waveStride) {
        const int p = t * 16 + lm;

        // Prefetch next tile's embedding rows (global_prefetch_b8; speculative OK)
        {
            int pn = p + waveStride * 16;
            __builtin_prefetch(e0 + pn, 0, 1);
            __builtin_prefetch(e1 + pn, 0, 1);
            __builtin_prefetch(e2 + pn, 0, 1);
        }

        const float x0 = e0[p];
        const float x1 = e1[p];
        const float x2 = e2[p];
        const float q  = i0 * x0 * x0 + i1 * x1 * x1 + i2 * x2 * x2;

        // B tile 4x16: lanes = N, VGPR0/1 = {K0,K1} (lanes 0-15), {K2,K3=0} (lanes 16-31)
        v2f B;
        B.x = half ? x2 : x0;
        B.y = half ? 0.0f : x1;

        // Fold q_p + b_n into the accumulator: C layout lane=N, VGPR j = row
        // (lanes 0-15: M=j ; lanes 16-31: M=j+8)
        v8f C[4];
#pragma unroll
        for (int g = 0; g < 4; ++g) {
            int mbase = g * 16 + half * 8;
#pragma unroll
            for (int j = 0; j < 8; ++j)
                C[g][j] = q + sBias[mbase + j];
        }

        // D = A x B + C   (4 independent WMMAs, no D->A/B hazard chain)
        C[0] = __builtin_amdgcn_wmma_f32_16x16x4_f32(false, A[0], false, B, (short)0, C[0], false, false);
        C[1] = __builtin_amdgcn_wmma_f32_16x16x4_f32(false, A[1], false, B, (short)0, C[1], false, false);
        C[2] = __builtin_amdgcn_wmma_f32_16x16x4_f32(false, A[2], false, B, (short)0, C[2], false, false);
        C[3] = __builtin_amdgcn_wmma_f32_16x16x4_f32(false, A[3], false, B, (short)0, C[3], false, false);

        // Epilogue: exp + coalesced stores (16 consecutive floats per half-wave per j)
#pragma unroll
        for (int g = 0; g < 4; ++g) {
            const int mbase = g * 16 + half * 8;
            float* __restrict__ ob = out + (size_t)mbase * (size_t)P + (size_t)(t * 16 + lm);
#pragma unroll
            for (int j = 0; j < 8; ++j) {
                ob[(size_t)j * (size_t)P] = __expf(C[g][j]);
            }
        }
    }
}

extern "C" void kernel_launch(void* const* d_in, const int* in_sizes, int n_in,
                              void* d_out, int out_size, void* d_ws, size_t ws_size,
                              hipStream_t stream) {
    const float* emb   = (const float*)d_in[0];   // [3,192,192,24]
    const float* cent  = (const float*)d_in[1];   // [64,3]
    const float* sigma = (const float*)d_in[2];   // [3]
    float* out = (float*)d_out;                   // [64,192,192,24]

    const int P = 192 * 192 * 24;                 // 884736 pixels
    const int numTiles = P / 16;                  // 55296 (exact)

    // 864 blocks x 8 waves = 6912 waves -> 8 tiles per wave (amortizes A/LDS setup)
    dim3 grid(864), block(256);
    embed2prob_wmma_kernel<<<grid, block, 0, stream>>>(emb, cent, sigma, out, P, numTiles);
}